// RBF_55095840473584
// MI455X (gfx1250) — compile-verified
//
#include <hip/hip_runtime.h>
#include <hip/hip_bf16.h>

typedef __attribute__((ext_vector_type(2))) float v2f;
typedef __attribute__((ext_vector_type(4))) float v4f;
typedef __attribute__((ext_vector_type(8))) float v8f;

#define N 8192
#define D 128
#define TILE 128
#define LDS_STRIDE 132   // 128 + 4 pad: row stride 4 banks -> conflict-free column reads

// ---------------------------------------------------------------------------
// Kernel 1: per-row squared norms. One wave32 per row; each lane loads float4.
// ---------------------------------------------------------------------------
__global__ __launch_bounds__(256) void rowsq_kernel(const float* __restrict__ X,
                                                    float* __restrict__ sq) {
    const int row  = blockIdx.x * 8 + (threadIdx.x >> 5);
    const int lane = threadIdx.x & 31;
    v4f v = *(const v4f*)(X + (size_t)row * D + lane * 4);
    float s = v[0] * v[0] + v[1] * v[1] + v[2] * v[2] + v[3] * v[3];
#pragma unroll
    for (int off = 16; off > 0; off >>= 1) s += __shfl_xor(s, off, 32);
    if (lane == 0) sq[row] = s;
}

// ---------------------------------------------------------------------------
// Kernel 2: deterministic per-block column-sum partials of X.
// 64 blocks x 256 threads; block b covers rows [b*128, b*128+128).
// ---------------------------------------------------------------------------
__global__ __launch_bounds__(256) void colsum_kernel(const float* __restrict__ X,
                                                     float* __restrict__ colpart) {
    __shared__ v4f part[256];
    const int tid = threadIdx.x;
    const int c   = tid & 31;           // column group (4 consecutive cols)
    const int w   = tid >> 5;           // row sub-group
    const int r0  = blockIdx.x * 128 + w * 16;

    v4f acc = {0.0f, 0.0f, 0.0f, 0.0f};
#pragma unroll
    for (int r = 0; r < 16; ++r) {
        v4f v = *(const v4f*)(X + (size_t)(r0 + r) * D + c * 4);
        acc += v;
    }
    part[tid] = acc;
    __syncthreads();
    if (tid < 32) {
        v4f s = part[c];                // group 0
#pragma unroll
        for (int g = 1; g < 8; ++g) s += part[g * 32 + c];
        *(v4f*)(colpart + blockIdx.x * D + c * 4) = s;
    }
}

// ---------------------------------------------------------------------------
// Kernel 3: analytic bandwidth, double precision, fixed-order reductions.
//   sum(l2) = 2*N*sum(sq) - 2*||sum_i x_i||^2   (clamp effect ~1e-10 rel.)
//   bw      = sum(l2) / (N*N - N)
// Emits 5 precomputed exp2-space coefficients:
//   coef[k] = -log2(e) / (bw * 2^(k-2))
// so the hot kernel does a bare v_exp_f32 per term (no mul by log2e, no div).
// ---------------------------------------------------------------------------
__global__ __launch_bounds__(256) void finalize_bw_kernel(const float* __restrict__ sq,
                                                          const float* __restrict__ colpart,
                                                          float* __restrict__ coef) {
    __shared__ double red[256];
    const int tid = threadIdx.x;

    double ssq = 0.0;
    for (int i = tid; i < N; i += 256) ssq += (double)sq[i];

    double sdot = 0.0;
    if (tid < D) {
        double s = 0.0;
        for (int b = 0; b < 64; ++b) s += (double)colpart[b * D + tid];
        sdot = s * s;
    }

    red[tid] = ssq;
    __syncthreads();
    for (int off = 128; off > 0; off >>= 1) {
        if (tid < off) red[tid] += red[tid + off];
        __syncthreads();
    }
    const double total_ssq = red[0];
    __syncthreads();

    red[tid] = sdot;
    __syncthreads();
    for (int off = 128; off > 0; off >>= 1) {
        if (tid < off) red[tid] += red[tid + off];
        __syncthreads();
    }
    if (tid == 0) {
        const double sum_l2 = 2.0 * (double)N * total_ssq - 2.0 * red[0];
        const double bw     = sum_l2 / ((double)N * (double)N - (double)N);
        const double LOG2E  = 1.4426950408889634073599246810019;
        const double base   = -LOG2E / bw;          // k=2 term
        coef[0] = (float)(base * 4.0);              // mult 0.25
        coef[1] = (float)(base * 2.0);              // mult 0.5
        coef[2] = (float)(base);                    // mult 1
        coef[3] = (float)(base * 0.5);              // mult 2
        coef[4] = (float)(base * 0.25);             // mult 4
    }
}

// ---------------------------------------------------------------------------
// Kernel 4: fused GEMM + RBF mixture. 128x128 output tile per block.
// WMMA f32 16x16x4 from LDS; epilogue computes
//   out = sum_k exp2(coef_k * max(0, sq_i + sq_j - 2*dot))
// and writes the final output ONCE (no l2 round-trip through HBM).
// ---------------------------------------------------------------------------
__global__ __launch_bounds__(256) void gemm_rbf_kernel(const float* __restrict__ X,
                                                       const float* __restrict__ sq,
                                                       const float* __restrict__ coef,
                                                       float* __restrict__ out) {
    __shared__ float As[TILE * LDS_STRIDE];
    __shared__ float Bs[TILE * LDS_STRIDE];

    const int tid = threadIdx.x;
    const int i0  = blockIdx.y * TILE;   // output rows  (A panel: X rows i0..)
    const int j0  = blockIdx.x * TILE;   // output cols  (B panel: X rows j0..)

    // Uniform precomputed exp2-space coefficients (scalar loads).
    const float c0 = coef[0];
    const float c1 = coef[1];
    const float c2 = coef[2];
    const float c3 = coef[3];
    const float c4 = coef[4];

    // Cooperative load: 128 rows x 128 floats per panel, float4 per lane-step.
#pragma unroll
    for (int it = 0; it < 16; ++it) {
        int idx = tid + it * 256;
        int r   = idx >> 5;
        int c   = (idx & 31) << 2;
        v4f a = *(const v4f*)(X + (size_t)(i0 + r) * D + c);
        v4f b = *(const v4f*)(X + (size_t)(j0 + r) * D + c);
        *(v4f*)(&As[r * LDS_STRIDE + c]) = a;
        *(v4f*)(&Bs[r * LDS_STRIDE + c]) = b;
    }
    __syncthreads();

    const int wave = tid >> 5;
    const int lane = tid & 31;
    const int half = lane >> 4;          // K sub-pair select per ISA layout
    const int l16  = lane & 15;
    const int wrow = (wave >> 1) * 32;   // wave covers 32 rows
    const int wcol = (wave & 1) * 64;    // wave covers 64 cols

    v8f acc[2][4] = {};

#pragma unroll 4
    for (int k0 = 0; k0 < D; k0 += 4) {
        const int kc = k0 + 2 * half;
        // A fragments: lane<16 -> {K=k0,k0+1}, lane>=16 -> {K=k0+2,k0+3}
        v2f a0 = *(const v2f*)(&As[(wrow +      l16) * LDS_STRIDE + kc]);
        v2f a1 = *(const v2f*)(&As[(wrow + 16 + l16) * LDS_STRIDE + kc]);
        // B fragments: B[k][n] = X_j[n][k]; same column-read pattern
        v2f b0 = *(const v2f*)(&Bs[(wcol +      l16) * LDS_STRIDE + kc]);
        v2f b1 = *(const v2f*)(&Bs[(wcol + 16 + l16) * LDS_STRIDE + kc]);
        v2f b2 = *(const v2f*)(&Bs[(wcol + 32 + l16) * LDS_STRIDE + kc]);
        v2f b3 = *(const v2f*)(&Bs[(wcol + 48 + l16) * LDS_STRIDE + kc]);

        acc[0][0] = __builtin_amdgcn_wmma_f32_16x16x4_f32(false, a0, false, b0, (short)0, acc[0][0], false, false);
        acc[0][1] = __builtin_amdgcn_wmma_f32_16x16x4_f32(false, a0, false, b1, (short)0, acc[0][1], false, false);
        acc[0][2] = __builtin_amdgcn_wmma_f32_16x16x4_f32(false, a0, false, b2, (short)0, acc[0][2], false, false);
        acc[0][3] = __builtin_amdgcn_wmma_f32_16x16x4_f32(false, a0, false, b3, (short)0, acc[0][3], false, false);
        acc[1][0] = __builtin_amdgcn_wmma_f32_16x16x4_f32(false, a1, false, b0, (short)0, acc[1][0], false, false);
        acc[1][1] = __builtin_amdgcn_wmma_f32_16x16x4_f32(false, a1, false, b1, (short)0, acc[1][1], false, false);
        acc[1][2] = __builtin_amdgcn_wmma_f32_16x16x4_f32(false, a1, false, b2, (short)0, acc[1][2], false, false);
        acc[1][3] = __builtin_amdgcn_wmma_f32_16x16x4_f32(false, a1, false, b3, (short)0, acc[1][3], false, false);
    }

    // Epilogue. C/D layout: VGPR r, lanes 0-15 -> M=r, lanes 16-31 -> M=r+8.
    // v_exp_f32 is a TRANS-pipe op and co-executes with the surrounding VALU.
#pragma unroll
    for (int mt = 0; mt < 2; ++mt) {
        const int ibase = i0 + wrow + mt * 16 + half * 8;
        float sqi[8];
#pragma unroll
        for (int r = 0; r < 8; ++r) sqi[r] = sq[ibase + r];
#pragma unroll
        for (int nt = 0; nt < 4; ++nt) {
            const int j     = j0 + wcol + nt * 16 + l16;
            const float sqj = sq[j];
#pragma unroll
            for (int r = 0; r < 8; ++r) {
                float v = sqi[r] + sqj - 2.0f * acc[mt][nt][r];
                v = fmaxf(v, 0.0f);
                float o = __builtin_amdgcn_exp2f(v * c0) +
                          __builtin_amdgcn_exp2f(v * c1) +
                          __builtin_amdgcn_exp2f(v * c2) +
                          __builtin_amdgcn_exp2f(v * c3) +
                          __builtin_amdgcn_exp2f(v * c4);
                out[(size_t)(ibase + r) * N + j] = o;
            }
        }
    }
}

// ---------------------------------------------------------------------------
extern "C" void kernel_launch(void* const* d_in, const int* in_sizes, int n_in,
                              void* d_out, int out_size, void* d_ws, size_t ws_size,
                              hipStream_t stream) {
    const float* X   = (const float*)d_in[0];
    float*       out = (float*)d_out;
    float*       ws  = (float*)d_ws;

    float* sq      = ws;                  // 8192 floats
    float* colpart = ws + N;              // 64*128 = 8192 floats
    float* coef    = ws + N + 64 * D;     // 5 floats

    rowsq_kernel<<<N / 8, 256, 0, stream>>>(X, sq);
    colsum_kernel<<<N / 128, 256, 0, stream>>>(X, colpart);
    finalize_bw_kernel<<<1, 256, 0, stream>>>(sq, colpart, coef);

    dim3 grid(N / TILE, N / TILE);        // 64 x 64 blocks
    gemm_rbf_kernel<<<grid, 256, 0, stream>>>(X, sq, coef, out);
}